// GroupSupConLoss_44186623541435
// MI455X (gfx1250) — compile-verified
//
#include <hip/hip_runtime.h>
#include <hip/hip_bf16.h>

// GroupSupConLoss fused on gfx1250:
//   sim = (E @ E^T)/tau via v_wmma_f32_16x16x4_f32, LDS double-buffered tiles
//   fed by GLOBAL_LOAD_ASYNC_TO_LDS_B128 (ASYNCcnt) when available,
//   fused row-wise shifted-exp sums + label-masked positive sums (f32 atomics),
//   epilogue kernel: label histogram -> num_pos, final scalar reduction.

typedef float v2f __attribute__((ext_vector_type(2)));
typedef float v8f __attribute__((ext_vector_type(8)));

#define TAU_INV   10.0f   // 1/tau
#define LSE_SHIFT 10.0f   // sim in [-10,10] for unit-norm embeddings

#define KC      32        // K-chunk staged per buffer
#define LSTRIDE 36        // padded row stride in floats (kills LDS bank conflicts)

#if defined(__gfx1250__) && __has_builtin(__builtin_amdgcn_global_load_async_to_lds_b128)
#define HAVE_ASYNC 1
typedef int v4i_gcc __attribute__((vector_size(16)));
typedef __attribute__((address_space(1))) v4i_gcc* gbl_v4p;   // prints as "__device__ *"
typedef __attribute__((address_space(3))) v4i_gcc* lds_v4p;   // prints as "__shared__ *"
static __device__ __forceinline__ void async_cp16(const void* g, void* l) {
    __builtin_amdgcn_global_load_async_to_lds_b128((gbl_v4p)g, (lds_v4p)l, 0, 0);
}
static __device__ __forceinline__ void wait_async_le(int n) {
#if __has_builtin(__builtin_amdgcn_s_wait_asynccnt)
    if (n == 0) __builtin_amdgcn_s_wait_asynccnt(0);
    else        __builtin_amdgcn_s_wait_asynccnt(8);
#else
    if (n == 0) asm volatile("s_wait_asynccnt 0x0" ::: "memory");
    else        asm volatile("s_wait_asynccnt 0x8" ::: "memory");
#endif
}
#else
#define HAVE_ASYNC 0
#endif

// ---------------------------------------------------------------- zero scratch
__global__ void zero_ws_kernel(float* __restrict__ p, int n) {
    int i = blockIdx.x * blockDim.x + threadIdx.x;
    if (i < n) p[i] = 0.0f;
}

// ---------------------------------------------------------------- main GEMM+reduce
// grid = (B/128, B/128), block = 256 threads = 8 wave32s.
// Workgroup tile: 128 rows x 128 cols. Wave w: rows [w*16, w*16+16) x 128 cols,
// 8 fp32 WMMA accumulators; A/B tiles staged in LDS, double buffered.
__global__ __launch_bounds__(256)
void supcon_tile_kernel(const float* __restrict__ E,
                        const int*   __restrict__ labels,
                        float* __restrict__ Ssum,
                        float* __restrict__ Psum,
                        int D)
{
    __shared__ float As[2][128 * LSTRIDE];
    __shared__ float Bs[2][128 * LSTRIDE];

    const int tid  = threadIdx.x;
    const int wave = tid >> 5;
    const int lane = tid & 31;
    const int lo   = lane & 15;   // M/N index within fragment
    const int hi   = lane >> 4;   // K-pair select

    const int rowTile = blockIdx.y * 128;
    const int colTile = blockIdx.x * 128;
    const int rowBase = rowTile + wave * 16;

    v8f acc[8];
#pragma unroll
    for (int c = 0; c < 8; ++c) acc[c] = (v8f){};

    // ---- staging: each thread moves 16 floats of A and 16 of B per chunk ----
    const int sr = tid >> 1;          // 0..127 : tile-local row
    const int sh = (tid & 1) << 4;    // 0 or 16 : half-chunk of K
    auto stage = [&](int buf, int k0) {
        const float* gA = E + (size_t)(rowTile + sr) * D + k0 + sh;
        const float* gB = E + (size_t)(colTile + sr) * D + k0 + sh;
        float* sA = &As[buf][sr * LSTRIDE + sh];
        float* sB = &Bs[buf][sr * LSTRIDE + sh];
#if HAVE_ASYNC
#pragma unroll
        for (int i = 0; i < 4; ++i) {
            async_cp16(gA + 4 * i, sA + 4 * i);   // 4 x b128 -> ASYNCcnt
            async_cp16(gB + 4 * i, sB + 4 * i);
        }
#else
#pragma unroll
        for (int i = 0; i < 4; ++i) ((float4*)sA)[i] = ((const float4*)gA)[i];
#pragma unroll
        for (int i = 0; i < 4; ++i) ((float4*)sB)[i] = ((const float4*)gB)[i];
#endif
    };

    // ---- compute one staged chunk: 8 K-steps x 8 col-tiles of WMMA ----
    auto compute = [&](int buf) {
        const float* sAw = &As[buf][(wave * 16 + lo) * LSTRIDE + 2 * hi];
        const float* sBl = &Bs[buf][lo * LSTRIDE + 2 * hi];
#pragma unroll
        for (int ks = 0; ks < KC / 4; ++ks) {
            v2f a = *(const v2f*)(sAw + ks * 4);
            v2f bf[8];
#pragma unroll
            for (int c = 0; c < 8; ++c)
                bf[c] = *(const v2f*)(sBl + c * (16 * LSTRIDE) + ks * 4);
#pragma unroll
            for (int c = 0; c < 8; ++c)
                acc[c] = __builtin_amdgcn_wmma_f32_16x16x4_f32(
                    false, a, false, bf[c], (short)0, acc[c], false, false);
        }
    };

    const int NC = D / KC;
    stage(0, 0);                       // prologue: fill buffer 0
    for (int ch = 0; ch < NC; ++ch) {
        const int buf = ch & 1;
        if (ch + 1 < NC) stage(buf ^ 1, (ch + 1) * KC);  // prefetch next chunk
#if HAVE_ASYNC
        wait_async_le((ch + 1 < NC) ? 8 : 0);            // current chunk landed
#endif
        __syncthreads();
        compute(buf);
        __syncthreads();               // buffer safe to overwrite next iter
    }

    // ---- fused epilogue: row-wise reductions ----
    int labc[8];
#pragma unroll
    for (int c = 0; c < 8; ++c) labc[c] = labels[colTile + 16 * c + lo];

#pragma unroll
    for (int v = 0; v < 8; ++v) {
        const int gr   = rowBase + v + 8 * hi;   // this half-wave's global row
        const int labR = labels[gr];

        float sExp = 0.0f, sPos = 0.0f;
#pragma unroll
        for (int c = 0; c < 8; ++c) {
            float s  = acc[c][v] * TAU_INV;
            int   gc = colTile + 16 * c + lo;
            bool  diag = (gc == gr);
            sExp += diag ? 0.0f : __expf(s - LSE_SHIFT);
            sPos += (!diag && (labc[c] == labR)) ? s : 0.0f;
        }
#pragma unroll
        for (int off = 1; off < 16; off <<= 1) {  // in-half 16-lane reduce
            sExp += __shfl_xor(sExp, off, 32);
            sPos += __shfl_xor(sPos, off, 32);
        }
        if (lo == 0) {
            atomicAdd(&Ssum[gr], sExp);
            atomicAdd(&Psum[gr], sPos);
        }
    }
}

// ---------------------------------------------------------------- final reduce
__global__ __launch_bounds__(1024)
void supcon_finalize_kernel(const int*   __restrict__ labels,
                            const float* __restrict__ Ssum,
                            const float* __restrict__ Psum,
                            float* __restrict__ out, int B)
{
    __shared__ int   hist[256];
    __shared__ float redS[32];
    __shared__ float redC[32];

    const int t = threadIdx.x;
    if (t < 256) hist[t] = 0;
    __syncthreads();
    for (int i = t; i < B; i += 1024) atomicAdd(&hist[labels[i] & 255], 1);
    __syncthreads();

    float lsum = 0.0f, lcnt = 0.0f;
    for (int i = t; i < B; i += 1024) {
        int np = hist[labels[i] & 255] - 1;
        if (np > 0) {
            float lse      = LSE_SHIFT + logf(Ssum[i]);
            float mean_pos = Psum[i] / (float)np;
            lsum += lse - mean_pos;
            lcnt += 1.0f;
        }
    }
#pragma unroll
    for (int off = 16; off; off >>= 1) {
        lsum += __shfl_down(lsum, off, 32);
        lcnt += __shfl_down(lcnt, off, 32);
    }
    const int w = t >> 5, l = t & 31;
    if (l == 0) { redS[w] = lsum; redC[w] = lcnt; }
    __syncthreads();
    if (t < 32) {
        lsum = redS[t];
        lcnt = redC[t];
#pragma unroll
        for (int off = 16; off; off >>= 1) {
            lsum += __shfl_down(lsum, off, 32);
            lcnt += __shfl_down(lcnt, off, 32);
        }
        if (t == 0) out[0] = lsum / fmaxf(lcnt, 1.0f);
    }
}

// ---------------------------------------------------------------- launch
extern "C" void kernel_launch(void* const* d_in, const int* in_sizes, int n_in,
                              void* d_out, int out_size, void* d_ws, size_t ws_size,
                              hipStream_t stream) {
    const float* E      = (const float*)d_in[0];
    const int*   labels = (const int*)d_in[1];
    float*       out    = (float*)d_out;

    const int B = in_sizes[1];               // 8192
    const int D = in_sizes[0] / B;           // 1024

    float* Ssum = (float*)d_ws;
    float* Psum = Ssum + B;

    zero_ws_kernel<<<(2 * B + 255) / 256, 256, 0, stream>>>(Ssum, 2 * B);

    dim3 grid(B / 128, B / 128);
    supcon_tile_kernel<<<grid, 256, 0, stream>>>(E, labels, Ssum, Psum, D);

    supcon_finalize_kernel<<<1, 1024, 0, stream>>>(labels, Ssum, Psum, out, B);
}